// MulticlassNMS_71451075936892
// MI455X (gfx1250) — compile-verified
//
#include <hip/hip_runtime.h>
#include <hip/hip_bf16.h>
#include <stdint.h>

// ---------------- problem constants (match reference) ----------------
#define N_BOX   5000
#define N_CLS   81
#define N_FG    80
#define WROW    157            // ceil(5000/32) mask words per row
#define MAXI    300
#define NEG_INF (-3.402823466e38f)

typedef float v2f __attribute__((ext_vector_type(2)));
typedef float v8f __attribute__((ext_vector_type(8)));
typedef unsigned int u32x4 __attribute__((ext_vector_type(4)));
typedef int i32x8 __attribute__((ext_vector_type(8)));
typedef int i32x4 __attribute__((ext_vector_type(4)));

#if defined(__has_builtin)
#if __has_builtin(__builtin_amdgcn_tensor_load_to_lds) && __has_builtin(__builtin_amdgcn_s_wait_tensorcnt)
#define HAVE_TDM 1
#endif
#endif
#ifndef HAVE_TDM
#define HAVE_TDM 0
#endif

// =====================================================================
// Kernel A: pairwise IoU-threshold bitmask. One wave32 per 16x32 tile.
// area_i + area_j computed with V_WMMA_F32_16X16X4_F32 (rank-2 update):
//   A(16x4) rows = [area_i, 1, 0, 0], B(4x16) cols = [1, area_j, 0, 0]^T
// Division-free threshold: inter*(1+t) > t*(area_i+area_j)
//   <=> inter/(ai+aj-inter) > t  (union > 0 always here).
// =====================================================================
__global__ __launch_bounds__(32)
void nms_mask_kernel(const float* __restrict__ bboxes,
                     const float* __restrict__ nms_thresh_p,
                     uint32_t* __restrict__ mask)
{
    const int lane = threadIdx.x;   // 0..31
    const int ln   = lane & 15;
    const int half = lane >> 4;
    const int i0   = blockIdx.y * 16;
    const int j0   = blockIdx.x * 32;
    const float nt = nms_thresh_p[0];

    const float4* bb = (const float4*)bboxes;

    // column boxes for two 16-wide column sub-tiles
    const int jc0 = min(j0 + ln,      N_BOX - 1);
    const int jc1 = min(j0 + 16 + ln, N_BOX - 1);
    float4 bj0 = bb[jc0];
    float4 bj1 = bb[jc1];
    float aj0 = (bj0.z - bj0.x) * (bj0.w - bj0.y);
    float aj1 = (bj1.z - bj1.x) * (bj1.w - bj1.y);

    const int ic = min(i0 + ln, N_BOX - 1);
    float4 bic = bb[ic];
    float ai = (bic.z - bic.x) * (bic.w - bic.y);

    // A-matrix 16x4 f32 layout: VGPR0 = {K0 | K2}, VGPR1 = {K1 | K3}, M = lane&15
    v2f a;  a.x  = half ? 0.f : ai;   a.y  = half ? 0.f : 1.f;
    // B-matrix 4x16 f32 (assumed mirrored layout), N = lane&15
    v2f b0; b0.x = half ? 0.f : 1.f;  b0.y = half ? 0.f : aj0;
    v2f b1; b1.x = half ? 0.f : 1.f;  b1.y = half ? 0.f : aj1;
    v8f cz = {};
    // D[m][n] = area_i[m] + area_j[n]
    v8f d0 = __builtin_amdgcn_wmma_f32_16x16x4_f32(false, a, false, b0, (short)0, cz, false, false);
    v8f d1 = __builtin_amdgcn_wmma_f32_16x16x4_f32(false, a, false, b1, (short)0, cz, false, false);

    // zero out bits for columns j >= N_BOX (clamped loads would alias row data)
    const uint32_t vmaskw = (N_BOX - j0 >= 32) ? 0xFFFFFFFFu
                          : ((N_BOX > j0) ? ((1u << (N_BOX - j0)) - 1u) : 0u);

    #pragma unroll
    for (int r = 0; r < 8; ++r) {
        const int m = half ? (r + 8) : r;          // C/D layout: VGPR r -> M=r (lo) / r+8 (hi)
        const int i = min(i0 + m, N_BOX - 1);
        float4 bi = bb[i];

        float iw0 = fminf(bi.z, bj0.z) - fmaxf(bi.x, bj0.x);
        float ih0 = fminf(bi.w, bj0.w) - fmaxf(bi.y, bj0.y);
        float in0 = fmaxf(iw0, 0.f) * fmaxf(ih0, 0.f);
        int   p0  = in0 * (1.f + nt) > nt * d0[r];

        float iw1 = fminf(bi.z, bj1.z) - fmaxf(bi.x, bj1.x);
        float ih1 = fminf(bi.w, bj1.w) - fmaxf(bi.y, bj1.y);
        float in1 = fmaxf(iw1, 0.f) * fmaxf(ih1, 0.f);
        int   p1  = in1 * (1.f + nt) > nt * d1[r];

        uint32_t m0 = (uint32_t)(__ballot(p0) & 0xFFFFFFFFull); // lo16: row r, hi16: row r+8 (cols j0..j0+15)
        uint32_t m1 = (uint32_t)(__ballot(p1) & 0xFFFFFFFFull); // same for cols j0+16..j0+31

        uint32_t wlo = ((m0 & 0xFFFFu) | ((m1 & 0xFFFFu) << 16)) & vmaskw;  // row i0+r
        uint32_t whi = ((m0 >> 16)     | (m1 & 0xFFFF0000u))     & vmaskw;  // row i0+r+8

        if (lane == 0) {
            int rlo = i0 + r, rhi = i0 + r + 8;
            if (rlo < N_BOX) mask[(size_t)rlo * WROW + blockIdx.x] = wlo;
            if (rhi < N_BOX) mask[(size_t)rhi * WROW + blockIdx.x] = whi;
        }
    }
}

// =====================================================================
// Kernel B: per-class greedy NMS via repeated block-argmax (equivalent to
// sorted sequential scan). One 256-thread workgroup per fg class.
// Class-score column is staged into LDS by the Tensor Data Mover:
// 2D tile 1x5000 with tensor_dim0_stride = 81 floats (strided gather).
// =====================================================================
__global__ __launch_bounds__(256)
void nms_class_kernel(const float* __restrict__ scores,
                      const float* __restrict__ conf_p,
                      const uint32_t* __restrict__ mask,
                      float* __restrict__ flat)
{
    __shared__ float    s[N_BOX];
    __shared__ float    rv[256];
    __shared__ int      ri[256];
    __shared__ uint32_t keepw[WROW];

    const int   tid = threadIdx.x;
    const int   c   = blockIdx.x;          // fg class -> scores column c+1
    const float ct  = conf_p[0];

#if HAVE_TDM
    if (tid < 32) {                        // wave 0 issues one TDM gather
        uint64_t gaddr = (uint64_t)(uintptr_t)(scores + (c + 1));
        uint32_t ldsa  = (uint32_t)(uintptr_t)&s[0];   // ISA 10.2: LDS address is the pointer's low 32 bits
        u32x4 g0;
        g0.x = 1u;                                       // count=1 (valid), user mode
        g0.y = ldsa;                                     // D#.lds_addr
        g0.z = (uint32_t)gaddr;                          // D#.global_addr[31:0]
        g0.w = (uint32_t)(gaddr >> 32) | (2u << 30);     // global_addr[56:32] | type=2
        i32x8 g1 = {0, 0, 0, 0, 0, 0, 0, 0};
        g1[0] = (2 << 16);                 // workgroup_mask=0, data_size=2 (4B)
        g1[1] = (N_CLS << 16);             // tensor_dim0 = 81 (bits 79:48 low half)
        g1[2] = (int)((uint32_t)N_BOX << 16); // tensor_dim1 = 5000 (bits 111:80 low half)
        g1[3] = (1 << 16);                 // tensor_dim1 hi=0 | tile_dim0 = 1
        g1[4] = N_BOX;                     // tile_dim1 = 5000, tile_dim2 = 0
        g1[5] = N_CLS;                     // tensor_dim0_stride = 81 (low 32 of 48)
        i32x4 gz = {0, 0, 0, 0};           // groups 2/3 unused (2D tensor)
        i32x8 gz8 = {0, 0, 0, 0, 0, 0, 0, 0};
        __builtin_amdgcn_tensor_load_to_lds(g0, g1, gz, gz, gz8, 0);
        __builtin_amdgcn_s_wait_tensorcnt(0);
    }
    for (int w = tid; w < WROW; w += 256) keepw[w] = 0u;
    __syncthreads();
    for (int b = tid; b < N_BOX; b += 256) {
        float v = s[b];
        s[b] = (v > ct) ? v : NEG_INF;
    }
#else
    for (int b = tid; b < N_BOX; b += 256) {
        float v = scores[(size_t)b * N_CLS + (c + 1)];
        s[b] = (v > ct) ? v : NEG_INF;
    }
    for (int w = tid; w < WROW; w += 256) keepw[w] = 0u;
#endif
    __syncthreads();

    for (int iter = 0; iter < N_BOX; ++iter) {
        // block argmax (ties -> larger index, mimicking reversed argsort)
        float bv = NEG_INF; int bi_ = -1;
        for (int b = tid; b < N_BOX; b += 256) {
            float v = s[b];
            if (v > bv || (v == bv && b > bi_)) { bv = v; bi_ = b; }
        }
        rv[tid] = bv; ri[tid] = bi_;
        __syncthreads();
        for (int st = 128; st > 0; st >>= 1) {
            if (tid < st) {
                float ov = rv[tid + st]; int oi = ri[tid + st];
                if (ov > rv[tid] || (ov == rv[tid] && oi > ri[tid])) { rv[tid] = ov; ri[tid] = oi; }
            }
            __syncthreads();
        }
        const float val = rv[0];
        const int   idx = ri[0];
        __syncthreads();
        if (val < -1.0e38f) break;          // uniform: nothing valid left

        if (tid == 0) {
            keepw[idx >> 5] |= (1u << (idx & 31));
            s[idx] = NEG_INF;
        }
        // suppress everything overlapping idx (diagonal bit kills idx too)
        const uint32_t* row = mask + (size_t)idx * WROW;
        __builtin_prefetch(row, 0, 0);      // -> global_prefetch_b8
        for (int w = tid; w < WROW; w += 256) {
            uint32_t bits = row[w];
            while (bits) {
                int bpos = __ffs(bits) - 1;
                bits &= bits - 1;
                int j = w * 32 + bpos;
                if (j < N_BOX) s[j] = NEG_INF;
            }
        }
        __syncthreads();
    }

    for (int b = tid; b < N_BOX; b += 256) {
        bool k = (keepw[b >> 5] >> (b & 31)) & 1u;
        flat[(size_t)c * N_BOX + b] = k ? scores[(size_t)b * N_CLS + (c + 1)] : NEG_INF;
    }
}

// =====================================================================
// Kernel C: sorted top-300 over the 400K flat scores (L2-resident) via
// 300 sequential block arg-maxes; writes preds (300x5) then labels (300).
// =====================================================================
__global__ __launch_bounds__(1024)
void topk_kernel(const float* __restrict__ bboxes,
                 const float* __restrict__ scores,
                 const int* __restrict__ max_inst_p,
                 float* __restrict__ flat,
                 float* __restrict__ out)
{
    __shared__ float rv[1024];
    __shared__ int   ri[1024];
    const int tid   = threadIdx.x;
    const int total = N_FG * N_BOX;
    int mi = max_inst_p[0]; if (mi > MAXI) mi = MAXI;

    for (int t = 0; t < MAXI; ++t) {
        float bv = NEG_INF; int bi_ = -1;
        for (int f = tid; f < total; f += 1024) {
            float v = flat[f];
            if (v > bv || (v == bv && f > bi_)) { bv = v; bi_ = f; }
        }
        rv[tid] = bv; ri[tid] = bi_;
        __syncthreads();
        for (int st = 512; st > 0; st >>= 1) {
            if (tid < st) {
                float ov = rv[tid + st]; int oi = ri[tid + st];
                if (ov > rv[tid] || (ov == rv[tid] && oi > ri[tid])) { rv[tid] = ov; ri[tid] = oi; }
            }
            __syncthreads();
        }
        const float val = rv[0];
        const int   idx = ri[0];
        __syncthreads();

        if (tid == 0) {
            bool valid = (val > -1.7014118e38f) && (t < mi);  // > neg_inf/2
            int box = idx % N_BOX;
            int cls = idx / N_BOX + 1;
            float4 bbv = ((const float4*)bboxes)[box];
            out[t * 5 + 0] = valid ? bbv.x : 0.f;
            out[t * 5 + 1] = valid ? bbv.y : 0.f;
            out[t * 5 + 2] = valid ? bbv.z : 0.f;
            out[t * 5 + 3] = valid ? bbv.w : 0.f;
            out[t * 5 + 4] = valid ? scores[(size_t)box * N_CLS + cls] : 0.f;
            out[MAXI * 5 + t] = valid ? (float)cls : -1.f;   // labels
            flat[idx] = NEG_INF;                             // exclude from next pass
        }
        __threadfence();
        __syncthreads();
    }
}

// =====================================================================
extern "C" void kernel_launch(void* const* d_in, const int* in_sizes, int n_in,
                              void* d_out, int out_size, void* d_ws, size_t ws_size,
                              hipStream_t stream) {
    const float* bboxes = (const float*)d_in[0];   // (5000,4) f32
    const float* scores = (const float*)d_in[1];   // (5000,81) f32
    const float* conf   = (const float*)d_in[2];   // scalar f32
    const float* nmst   = (const float*)d_in[3];   // scalar f32
    const int*   maxi   = (const int*)d_in[4];     // scalar i32

    uint32_t* mask = (uint32_t*)d_ws;                                   // 5000*157*4  = ~3.1 MB
    float*    flat = (float*)((char*)d_ws + (size_t)N_BOX * WROW * 4);  // 80*5000*4   = ~1.6 MB
    float*    out  = (float*)d_out;                                     // 300*5 + 300

    dim3 gA(WROW, (N_BOX + 15) / 16);   // 157 x 313 tiles, 1 wave each
    nms_mask_kernel<<<gA, 32, 0, stream>>>(bboxes, nmst, mask);
    nms_class_kernel<<<N_FG, 256, 0, stream>>>(scores, conf, mask, flat);
    topk_kernel<<<1, 1024, 0, stream>>>(bboxes, scores, maxi, flat, out);
}